// SumAggregator_66245575573682
// MI455X (gfx1250) — compile-verified
//
#include <hip/hip_runtime.h>
#include <hip/hip_bf16.h>

typedef __attribute__((ext_vector_type(2))) float v2f;
typedef __attribute__((ext_vector_type(4))) float v4f;
typedef __attribute__((ext_vector_type(8))) float v8f;

#define IN_CH 256
#define OUT_CH 128
#define LDS_STRIDE 260   // 256 + 4 floats pad -> conflict-free ds_load_b64, 16B-aligned rows

// Fire-and-forget fp32 global atomic add (no return -> STOREcnt, L2 atomic units).
__device__ __forceinline__ void global_atomic_fadd(float* p, float v) {
  asm volatile("global_atomic_add_f32 %0, %1, off" :: "v"(p), "v"(v) : "memory");
}

// ---------------------------------------------------------------------------
// GEMM: h[n][c] = sum_k x[n][k] * W[c][k] + b[c]
// block = 256 threads = 8 waves. Block -> 16-node M tile; wave w -> N tile
// [16w, 16w+16). K looped 4 at a time with V_WMMA_F32_16X16X4_F32.
// x tile staged to LDS via async global->LDS copies (ASYNCcnt path).
// ---------------------------------------------------------------------------
__global__ __launch_bounds__(256) void gemm16x16x4_f32_kernel(
    const float* __restrict__ x, const float* __restrict__ W,
    const float* __restrict__ bias, float* __restrict__ h, int nNodes) {
  __shared__ float xs[16 * LDS_STRIDE];

  const int tid   = threadIdx.x;
  const int mbase = blockIdx.x * 16;
  const bool fullTile = (mbase + 16 <= nNodes);   // block-uniform

  if (fullTile) {
    // Async stage: 1024 x b128, 4 per lane. Per-lane global->LDS DMA, no VGPR bounce.
    #pragma unroll
    for (int i = 0; i < 4; ++i) {
      int idx = tid + i * 256;     // float4 index within the 16x256 tile
      int row = idx >> 6;          // 64 float4 per row
      int c4  = idx & 63;
      const float* gp = x + (size_t)(mbase + row) * IN_CH + (size_t)c4 * 4;
      unsigned lds_off = (unsigned)(uintptr_t)(&xs[row * LDS_STRIDE + c4 * 4]);
      asm volatile("global_load_async_to_lds_b128 %0, %1, off"
                   :: "v"(lds_off), "v"(gp) : "memory");
    }
    asm volatile("s_wait_asynccnt 0x0" ::: "memory");
  } else {
    // Tail tile: guarded synchronous stage with zero fill.
    #pragma unroll
    for (int i = 0; i < 4; ++i) {
      int idx = tid + i * 256;
      int row = idx >> 6;
      int c4  = idx & 63;
      int node = mbase + row;
      v4f val = {0.0f, 0.0f, 0.0f, 0.0f};
      if (node < nNodes)
        val = *(const v4f*)(x + (size_t)node * IN_CH + (size_t)c4 * 4);
      *(v4f*)(&xs[row * LDS_STRIDE + c4 * 4]) = val;
    }
  }
  __syncthreads();

  const int wave = tid >> 5;
  const int lane = tid & 31;
  const int half = lane >> 4;   // 0: lanes 0-15, 1: lanes 16-31
  const int l    = lane & 15;
  const int ch   = wave * 16 + l;          // output channel (N index)

  // A lanes: M = l, K = {2*half, 2*half+1} within each k-step of 4.
  // B lanes: N = l, K = {2*half, 2*half+1} -> contiguous in W[ch][*].
  const float* wp = W + (size_t)ch * IN_CH + 2 * half;
  const float* ap = &xs[l * LDS_STRIDE + 2 * half];

  v8f acc = {};
  #pragma unroll 8
  for (int k0 = 0; k0 < IN_CH; k0 += 4) {
    v2f a = *(const v2f*)(ap + k0);   // ds_load_b64 (2addr pairs)
    v2f b = *(const v2f*)(wp + k0);   // global_load_b64
    acc = __builtin_amdgcn_wmma_f32_16x16x4_f32(
        /*neg_a=*/false, a, /*neg_b=*/false, b,
        /*c_mod=*/(short)0, acc, /*reuse_a=*/false, /*reuse_b=*/false);
  }

  const float bv = bias[ch];
  // D layout: VGPR r -> M = r + 8*half, N = l (16x16 f32 C/D map).
  if (fullTile) {
    float* hp = h + (size_t)(mbase + half * 8) * OUT_CH + ch;
    #pragma unroll
    for (int r = 0; r < 8; ++r)
      hp[(size_t)r * OUT_CH] = acc[r] + bv;   // immediate-offset b32 stores
  } else {
    #pragma unroll
    for (int r = 0; r < 8; ++r) {
      int node = mbase + r + half * 8;
      if (node < nNodes)
        h[(size_t)node * OUT_CH + ch] = acc[r] + bv;
    }
  }
}

// ---------------------------------------------------------------------------
// Zero-init the output (harness poisons d_out with 0xAA).
// ---------------------------------------------------------------------------
__global__ void zero_kernel(float* __restrict__ out, int n) {
  int i = blockIdx.x * blockDim.x + threadIdx.x;
  if (i < n) out[i] = 0.0f;
}

// ---------------------------------------------------------------------------
// Edge scatter: one wave per edge; 32 lanes x float4 = 128 channels.
// out[dst][c] += w_e * h[src][c] via native fp32 global atomics (L2-resident).
// ---------------------------------------------------------------------------
__global__ __launch_bounds__(256) void scatter_kernel(
    const float* __restrict__ h, const int* __restrict__ esrc,
    const int* __restrict__ edst, const float* __restrict__ ew,
    float* __restrict__ out, int nEdges) {
  const int wavesPerBlock = blockDim.x >> 5;
  const int e = blockIdx.x * wavesPerBlock + (threadIdx.x >> 5);
  if (e >= nEdges) return;
  const int lane = threadIdx.x & 31;

  // e is wave-uniform: pin it so the edge scalars become SMEM loads.
  const int eu = __builtin_amdgcn_readfirstlane(e);
  const int   s = esrc[eu];
  const int   d = edst[eu];
  const float w = ew[eu];

  v4f v = *(const v4f*)(h + (size_t)s * OUT_CH + (size_t)lane * 4);
  float* op = out + (size_t)d * OUT_CH + (size_t)lane * 4;
  global_atomic_fadd(op + 0, v.x * w);
  global_atomic_fadd(op + 1, v.y * w);
  global_atomic_fadd(op + 2, v.z * w);
  global_atomic_fadd(op + 3, v.w * w);
}

// ---------------------------------------------------------------------------
extern "C" void kernel_launch(void* const* d_in, const int* in_sizes, int n_in,
                              void* d_out, int out_size, void* d_ws, size_t ws_size,
                              hipStream_t stream) {
  const float* x    = (const float*)d_in[0];
  const int*   esrc = (const int*)d_in[1];
  const int*   edst = (const int*)d_in[2];
  const float* ew   = (const float*)d_in[3];
  const float* W    = (const float*)d_in[4];
  const float* bias = (const float*)d_in[5];
  float* out = (float*)d_out;
  float* h   = (float*)d_ws;   // [nNodes x OUT_CH] f32 scratch (51.2 MB)

  const int nNodes = in_sizes[0] / IN_CH;
  const int nE     = in_sizes[1];

  const int mtiles = (nNodes + 15) / 16;
  gemm16x16x4_f32_kernel<<<dim3(mtiles), dim3(256), 0, stream>>>(x, W, bias, h, nNodes);

  zero_kernel<<<dim3((out_size + 255) / 256), dim3(256), 0, stream>>>(out, out_size);

  const int wavesPerBlock = 8;  // 256 threads
  scatter_kernel<<<dim3((nE + wavesPerBlock - 1) / wavesPerBlock), dim3(256), 0, stream>>>(
      h, esrc, edst, ew, out, nE);
}